// VoxelPooling_64209761076082
// MI455X (gfx1250) — compile-verified
//
#include <hip/hip_runtime.h>
#include <hip/hip_bf16.h>
#include <float.h>

// Problem constants (match reference)
#define BATCH   8
#define NPTS    65536
#define NCH     128
#define GRID    32
#define NVOX    32768           // GRID^3
#define GRANGE  3.2f
#define VSIZE   0.2f

#define PTS_PER_WAVE    32
#define WAVES_PER_BLOCK 8

// ---------------------------------------------------------------------------
// helpers
// ---------------------------------------------------------------------------
__device__ __forceinline__ int voxel_of(const float* __restrict__ pts, int pt) {
    float x = pts[(size_t)pt * 3 + 0];
    float y = pts[(size_t)pt * 3 + 1];
    float z = pts[(size_t)pt * 3 + 2];
    int cx = (int)((x + GRANGE) / VSIZE);   // keep real division: match reference numerics
    int cy = (int)((y + GRANGE) / VSIZE);
    int cz = (int)((z + GRANGE) / VSIZE);
    cx = min(max(cx, 0), GRID - 1);
    cy = min(max(cy, 0), GRID - 1);
    cz = min(max(cz, 0), GRID - 1);
    return cx + cy * GRID + cz * GRID * GRID;
}

// CDNA5 async global->LDS copy (ASYNCcnt-tracked DMA, 16B per lane)
__device__ __forceinline__ void async_load_b128(unsigned lds_byte_addr,
                                                const float* gaddr) {
    asm volatile("global_load_async_to_lds_b128 %0, %1, off"
                 :: "v"(lds_byte_addr), "v"(gaddr)
                 : "memory");
}

// Native no-return fmax atomic (global_atomic_max_num_f32).
// Inputs are NaN/denorm-free, so semantics match the reference segment_max.
__device__ __forceinline__ void atomic_fmax(float* addr, float v) {
    unsafeAtomicMax(addr, v);
}

// ---------------------------------------------------------------------------
// pass 0: zero per-voxel counts
// ---------------------------------------------------------------------------
__global__ void vp_zero_counts(int* __restrict__ cnt) {
    int i = blockIdx.x * blockDim.x + threadIdx.x;
    if (i < BATCH * NVOX) cnt[i] = 0;
}

// ---------------------------------------------------------------------------
// pass 1: per-voxel point counts + per-point global voxel id (-1 if masked)
// ---------------------------------------------------------------------------
__global__ void vp_count(const float* __restrict__ pts,
                         const unsigned char* __restrict__ mask,
                         int* __restrict__ cnt,
                         int* __restrict__ vidx) {
    int pt = blockIdx.x * blockDim.x + threadIdx.x;
    if (pt >= BATCH * NPTS) return;
    int gid = -1;
    if (mask[pt]) {
        int b = pt >> 16;             // NPTS = 65536
        gid = b * NVOX + voxel_of(pts, pt);
        atomicAdd(&cnt[gid], 1);
    }
    vidx[pt] = gid;
}

// ---------------------------------------------------------------------------
// pass 2: per-batch exclusive scan of occupancy -> compaction map + bool mask
// one 1024-thread block per batch; each thread owns 32 voxels
// ---------------------------------------------------------------------------
__global__ void vp_scan(const int* __restrict__ cnt,
                        int* __restrict__ pos,
                        int* __restrict__ totals,
                        float* __restrict__ mask_out) {
    __shared__ int sdata[1024];
    __shared__ int total_s;
    const int ELEMS = NVOX / 1024;    // 32
    int b   = blockIdx.x;
    int tid = threadIdx.x;
    int base = b * NVOX + tid * ELEMS;

    int local[ELEMS];
    int s = 0;
    #pragma unroll
    for (int k = 0; k < ELEMS; ++k) {
        int f = cnt[base + k] > 0;
        local[k] = s;                 // within-thread exclusive prefix
        s += f;
    }

    // Hillis-Steele inclusive scan across the 1024 threads
    sdata[tid] = s;
    __syncthreads();
    for (int off = 1; off < 1024; off <<= 1) {
        int t = (tid >= off) ? sdata[tid - off] : 0;
        __syncthreads();
        sdata[tid] += t;
        __syncthreads();
    }
    int excl = sdata[tid] - s;
    if (tid == 1023) total_s = sdata[1023];
    __syncthreads();
    int total = total_s;

    #pragma unroll
    for (int k = 0; k < ELEMS; ++k)
        pos[base + k] = excl + local[k];

    int mbase = tid * ELEMS;
    #pragma unroll
    for (int k = 0; k < ELEMS; ++k)
        mask_out[b * NVOX + mbase + k] = (mbase + k < total) ? 1.0f : 0.0f;

    if (tid == 0) totals[b] = total;
}

// ---------------------------------------------------------------------------
// pass 3: init output feature rows: -FLT_MAX for occupied slots, 0 for padding
// one thread per float4 (B*NVOX*32 threads)
// ---------------------------------------------------------------------------
__global__ void vp_init_feat(float4* __restrict__ out,
                             const int* __restrict__ totals) {
    int i = blockIdx.x * blockDim.x + threadIdx.x;
    const int TOTAL4 = BATCH * NVOX * (NCH / 4);
    if (i >= TOTAL4) return;
    int row = i / (NCH / 4);
    int b   = row / NVOX;
    int j   = row - b * NVOX;
    float v = (j < totals[b]) ? -FLT_MAX : 0.0f;
    float4 f; f.x = v; f.y = v; f.z = v; f.w = v;
    out[i] = f;
}

// ---------------------------------------------------------------------------
// pass 4: scatter-max. One wave per 32 points, one at a time; lane l owns channels [4l, 4l+4).
// Feature rows are DMA'd into LDS with the CDNA5 async-load path, double
// buffered so the DMA of row i+1 overlaps the atomics for row i. The voxel
// id is precomputed (vidx), so the inner loop is just:
//   async DMA -> s_wait_asynccnt -> ds_load_b128 -> pos lookup -> 4x fmax atomic
// ---------------------------------------------------------------------------
__global__ void vp_scatter(const int* __restrict__ vidx,
                           const float* __restrict__ feat,
                           const int* __restrict__ pos,
                           float* __restrict__ out) {
    __shared__ float smem[WAVES_PER_BLOCK][2][NCH];
    int lane = threadIdx.x & 31;
    int wib  = threadIdx.x >> 5;
    int wave = blockIdx.x * WAVES_PER_BLOCK + wib;
    int pt0  = wave * PTS_PER_WAVE;

    float* buf0 = &smem[wib][0][0];
    float* buf1 = &smem[wib][1][0];

    // prologue: DMA first row into buffer 0
    {
        const float* g = feat + (size_t)pt0 * NCH + lane * 4;
        unsigned l = (unsigned)(size_t)(buf0 + lane * 4);
        async_load_b128(l, g);
    }

    for (int i = 0; i < PTS_PER_WAVE; ++i) {
        int pt = pt0 + i;
        if (i + 1 < PTS_PER_WAVE) {
            // DMA next row into the other buffer, then wait for current row
            const float* g = feat + (size_t)(pt + 1) * NCH + lane * 4;
            float* dbuf = ((i + 1) & 1) ? buf1 : buf0;
            unsigned l = (unsigned)(size_t)(dbuf + lane * 4);
            async_load_b128(l, g);
            asm volatile("s_wait_asynccnt 0x1" ::: "memory");
        } else {
            asm volatile("s_wait_asynccnt 0x0" ::: "memory");
        }

        const float* rbuf = (i & 1) ? buf1 : buf0;
        float4 f = ((const float4*)rbuf)[lane];   // ds_load_b128

        int gid = vidx[pt];                       // same addr across wave
        if (gid >= 0) {                           // wave-uniform branch
            int dest = pos[gid];
            int b = pt >> 16;
            float* orow = out + ((size_t)b * NVOX + dest) * NCH + lane * 4;
            atomic_fmax(orow + 0, f.x);
            atomic_fmax(orow + 1, f.y);
            atomic_fmax(orow + 2, f.z);
            atomic_fmax(orow + 3, f.w);
        }
    }
}

// ---------------------------------------------------------------------------
// launch
// ---------------------------------------------------------------------------
extern "C" void kernel_launch(void* const* d_in, const int* in_sizes, int n_in,
                              void* d_out, int out_size, void* d_ws, size_t ws_size,
                              hipStream_t stream) {
    const float*         pts  = (const float*)d_in[0];
    const float*         feat = (const float*)d_in[1];
    const unsigned char* mask = (const unsigned char*)d_in[2];

    float* out_feat = (float*)d_out;                              // [B, NVOX, C]
    float* out_mask = out_feat + (size_t)BATCH * NVOX * NCH;      // [B, NVOX]

    int* cnt    = (int*)d_ws;                 // B*NVOX ints
    int* pos    = cnt + BATCH * NVOX;         // B*NVOX ints
    int* totals = pos + BATCH * NVOX;         // BATCH ints
    int* vidx   = totals + BATCH;             // B*NPTS ints

    vp_zero_counts<<<(BATCH * NVOX + 255) / 256, 256, 0, stream>>>(cnt);
    vp_count<<<(BATCH * NPTS + 255) / 256, 256, 0, stream>>>(pts, mask, cnt, vidx);
    vp_scan<<<BATCH, 1024, 0, stream>>>(cnt, pos, totals, out_mask);
    vp_init_feat<<<(BATCH * NVOX * (NCH / 4) + 255) / 256, 256, 0, stream>>>(
        (float4*)out_feat, totals);

    const int scatter_blocks = (BATCH * NPTS) / (PTS_PER_WAVE * WAVES_PER_BLOCK); // 2048
    vp_scatter<<<scatter_blocks, WAVES_PER_BLOCK * 32, 0, stream>>>(
        vidx, feat, pos, out_feat);
}